// Downsample_29274497089732
// MI455X (gfx1250) — compile-verified
//
#include <hip/hip_runtime.h>

typedef float v2f __attribute__((ext_vector_type(2)));
typedef float v4f __attribute__((ext_vector_type(4)));
typedef float v8f __attribute__((ext_vector_type(8)));

#define N_IN   2000000
#define N_OUT  250000
#define K_NB   8
#define C_IN   32
#define C_OUT  64
#define KTOT   256            // K_NB * C_IN
#define BN_EPS 1e-5f

#define ROWS_PER_BLOCK 64     // 4 waves x 16 rows
#define NBLK ((N_OUT + ROWS_PER_BLOCK - 1) / ROWS_PER_BLOCK)   // 3907

// ---------------------------------------------------------------------------
// Kernel 1: gathered GEMM (f32 WMMA 16x16x4) + bias, emit pre-BN output and
// per-block per-column {sum, sumsq} partials for the BatchNorm statistics.
// ---------------------------------------------------------------------------
__global__ __launch_bounds__(128) void octconv_gemm(
    const float* __restrict__ data,     // [N_IN, 32]
    const float* __restrict__ weight,   // [256, 64]
    const float* __restrict__ bias,     // [64]
    const int*   __restrict__ neigh,    // [N_OUT, 8]
    float*       __restrict__ out,      // [N_OUT, 64] (pre-BN here)
    float*       __restrict__ partial)  // [NBLK, 128] (sum | sumsq)
{
    __shared__ float ldsB[KTOT * C_OUT];          // 64 KB weight tile

    const int tid  = threadIdx.x;
    const int lane = tid & 31;
    const int wid  = tid >> 5;                    // rowTile 0..3
    const int ln   = lane & 15;
    const int kb   = (lane >> 4) << 1;            // 0 (lanes 0-15) / 2 (16-31)
    const int row0 = blockIdx.x * ROWS_PER_BLOCK;

    // Stage weights [K][N] into LDS (coalesced b128).
    for (int i = tid * 4; i < KTOT * C_OUT; i += 128 * 4)
        *(v4f*)&ldsB[i] = *(const v4f*)&weight[i];

    // This lane supplies A elements of gathered row (row0 + wid*16 + ln).
    const int arow = row0 + wid * 16 + ln;
    int nbr[K_NB];
    if (arow < N_OUT) {
#pragma unroll
        for (int j = 0; j < K_NB; ++j) nbr[j] = neigh[(size_t)arow * K_NB + j];
    } else {
#pragma unroll
        for (int j = 0; j < K_NB; ++j) nbr[j] = -1;
    }

    __syncthreads();

    v8f acc0 = {0.f,0.f,0.f,0.f,0.f,0.f,0.f,0.f};
    v8f acc1 = acc0, acc2 = acc0, acc3 = acc0;

#pragma unroll
    for (int j = 0; j < K_NB; ++j) {
        const int nbj = nbr[j];
        const float* __restrict__ ap =
            data + (size_t)(nbj < 0 ? 0 : nbj) * C_IN + kb;
#pragma unroll
        for (int kc = 0; kc < C_IN; kc += 4) {
            // A fragment: 16x4 f32, lane m holds K = kb..kb+1 of its row.
            v2f a = *(const v2f*)(ap + kc);
            if (nbj < 0) { a.x = 0.f; a.y = 0.f; }   // empty child -> zero
            const int k0 = j * C_IN + kc;
            const float* bp = &ldsB[(k0 + kb) * C_OUT + ln];
            v2f b0, b1, b2, b3;
            b0.x = bp[0];  b0.y = bp[0  + C_OUT];
            b1.x = bp[16]; b1.y = bp[16 + C_OUT];
            b2.x = bp[32]; b2.y = bp[32 + C_OUT];
            b3.x = bp[48]; b3.y = bp[48 + C_OUT];
            acc0 = __builtin_amdgcn_wmma_f32_16x16x4_f32(false, a, false, b0,
                                                         (short)0, acc0, false, false);
            acc1 = __builtin_amdgcn_wmma_f32_16x16x4_f32(false, a, false, b1,
                                                         (short)0, acc1, false, false);
            acc2 = __builtin_amdgcn_wmma_f32_16x16x4_f32(false, a, false, b2,
                                                         (short)0, acc2, false, false);
            acc3 = __builtin_amdgcn_wmma_f32_16x16x4_f32(false, a, false, b3,
                                                         (short)0, acc3, false, false);
        }
    }

    // Epilogue: +bias, store pre-BN, accumulate per-column sum / sumsq.
    // D layout: reg r -> row (r) for lanes 0-15, row (8+r) for lanes 16-31.
    const int halfOff = (lane >> 4) << 3;
    float s[4] = {0.f,0.f,0.f,0.f};
    float q[4] = {0.f,0.f,0.f,0.f};
#pragma unroll
    for (int nt = 0; nt < 4; ++nt) {
        const float bn = bias[nt * 16 + ln];
        v8f acc = (nt == 0) ? acc0 : (nt == 1) ? acc1 : (nt == 2) ? acc2 : acc3;
#pragma unroll
        for (int r = 0; r < 8; ++r) {
            const int grow = row0 + wid * 16 + halfOff + r;
            if (grow < N_OUT) {
                const float y = acc[r] + bn;
                out[(size_t)grow * C_OUT + nt * 16 + ln] = y;
                s[nt] += y;
                q[nt] += y * y;
            }
        }
    }

    __syncthreads();                 // done reading ldsB -> reuse for reduction
#pragma unroll
    for (int nt = 0; nt < 4; ++nt) {
        ldsB[(wid * 4 + nt) * 32 + lane]       = s[nt];
        ldsB[512 + (wid * 4 + nt) * 32 + lane] = q[nt];
    }
    __syncthreads();

    if (tid < C_OUT) {
        const int n = tid, nt = n >> 4, l2 = n & 15;
        float S = 0.f, Q = 0.f;
#pragma unroll
        for (int w = 0; w < 4; ++w) {
            S += ldsB[(w * 4 + nt) * 32 + l2] + ldsB[(w * 4 + nt) * 32 + l2 + 16];
            Q += ldsB[512 + (w * 4 + nt) * 32 + l2] +
                 ldsB[512 + (w * 4 + nt) * 32 + l2 + 16];
        }
        partial[(size_t)blockIdx.x * 128 + n]      = S;
        partial[(size_t)blockIdx.x * 128 + 64 + n] = Q;
    }
}

// ---------------------------------------------------------------------------
// Kernel 2: reduce block partials (fixed order -> deterministic), build the
// per-column affine coefficients  y' = a*y + b  with a = gamma*rstd,
// b = beta - mean*a.
// ---------------------------------------------------------------------------
__global__ __launch_bounds__(64) void bn_stats(
    const float* __restrict__ partial,
    const float* __restrict__ gamma,
    const float* __restrict__ beta,
    float*       __restrict__ coef)     // [128]: a | b
{
    const int n = threadIdx.x;          // 0..63
    float S = 0.f, Q = 0.f;
    for (int b = 0; b < NBLK; ++b) {
        S += partial[(size_t)b * 128 + n];
        Q += partial[(size_t)b * 128 + 64 + n];
    }
    const float inv  = 1.0f / (float)N_OUT;
    const float mean = S * inv;
    float var = Q * inv - mean * mean;
    var = var < 0.f ? 0.f : var;
    const float a = gamma[n] / sqrtf(var + BN_EPS);
    coef[n]      = a;
    coef[64 + n] = beta[n] - mean * a;
}

// ---------------------------------------------------------------------------
// Kernel 3: in-place elementwise BN apply, float4 vectorized.
// ---------------------------------------------------------------------------
__global__ __launch_bounds__(256) void bn_apply(
    float* __restrict__ out, const float* __restrict__ coef)
{
    __shared__ float sc[128];
    if (threadIdx.x < 128) sc[threadIdx.x] = coef[threadIdx.x];
    __syncthreads();

    const size_t gi   = (size_t)blockIdx.x * 256 + threadIdx.x;
    const size_t base = gi * 4;
    if (base < (size_t)N_OUT * C_OUT) {
        v4f v = *(v4f*)&out[base];
        const int c0 = (int)(base & 63);
        v.x = v.x * sc[c0]     + sc[64 + c0];
        v.y = v.y * sc[c0 + 1] + sc[64 + c0 + 1];
        v.z = v.z * sc[c0 + 2] + sc[64 + c0 + 2];
        v.w = v.w * sc[c0 + 3] + sc[64 + c0 + 3];
        *(v4f*)&out[base] = v;
    }
}

// ---------------------------------------------------------------------------
extern "C" void kernel_launch(void* const* d_in, const int* in_sizes, int n_in,
                              void* d_out, int out_size, void* d_ws, size_t ws_size,
                              hipStream_t stream)
{
    const float* data   = (const float*)d_in[0];   // [2M, 32]
    const float* weight = (const float*)d_in[1];   // [256, 64]
    const float* bias   = (const float*)d_in[2];   // [64]
    const float* gamma  = (const float*)d_in[3];   // [64]
    const float* beta   = (const float*)d_in[4];   // [64]
    const int*   neigh  = (const int*)d_in[5];     // [250k, 8]
    float* out = (float*)d_out;                    // [250k, 64]

    float* partial = (float*)d_ws;                 // NBLK*128 floats (~2 MB)
    float* coef    = partial + (size_t)NBLK * 128; // 128 floats

    octconv_gemm<<<NBLK, 128, 0, stream>>>(data, weight, bias, neigh, out, partial);
    bn_stats<<<1, 64, 0, stream>>>(partial, gamma, beta, coef);

    const int total4 = (N_OUT * C_OUT) / 4;        // 4,000,000
    bn_apply<<<(total4 + 255) / 256, 256, 0, stream>>>(out, coef);
}